// brain_gaussian_15101105013366
// MI455X (gfx1250) — compile-verified
//
#include <hip/hip_runtime.h>
#include <cstdint>

// Fused GaussianBlur(sigma=1, 7-tap, reflect-101 over H,W) + 0.05*noise for
// (C,H,W,D,T) = (2,96,96,64,20) float32; dt = D*T = 1280 contiguous.
//
// Memory-bound: 283 MB min traffic -> ~12 us roofline @ 23.3 TB/s.
// CDNA5 path: double-buffered Tensor-Data-Mover DMAs (tensor_load_to_lds,
// TENSORcnt <= 1 counted waits) stage clipped 3D halo windows (dt x w x h)
// into LDS while the previous chunk is blurred out of LDS with float4 math.
// Reflection handled at read time via small LDS index maps; img hits HBM once.

typedef float v4f __attribute__((ext_vector_type(4)));

namespace {

constexpr int Hh  = 96;
constexpr int Ww  = 96;
constexpr int DT  = 64 * 20;      // 1280
constexpr int TH  = 16;
constexpr int TW  = 16;
constexpr int DTC = 16;           // dt elements per chunk
constexpr int NCHUNK = 4;         // dt chunks per block (pipelined)
constexpr int TIH = TH + 6;       // 22 window rows (h)
constexpr int TIW = TW + 6;       // 22 window cols (w)
constexpr int NTHREADS = 256;     // 8 waves (wave32)

__device__ __forceinline__ int refl101(int i, int n) {
    return i < 0 ? -i : (i >= n ? 2 * n - 2 - i : i);
}

#if __has_builtin(__builtin_amdgcn_tensor_load_to_lds)
// Issue one TDM DMA: 3D window (16 dt x wcount x hcount) global -> LDS.
__device__ __forceinline__ void tdm_issue(const float* gsrc, uint32_t lds,
                                          int wcount, int hcount) {
    typedef unsigned int v4u __attribute__((ext_vector_type(4)));
    typedef int v8i __attribute__((ext_vector_type(8)));
    typedef int v4i __attribute__((ext_vector_type(4)));

    const uint64_t ga = (uint64_t)(uintptr_t)gsrc;

    v4u g0;
    g0[0] = 1u;                                   // count=1 (valid user D#)
    g0[1] = lds;                                  // lds_addr (bytes)
    g0[2] = (uint32_t)ga;                         // global_addr[31:0]
    g0[3] = (uint32_t)((ga >> 32) & 0x01FFFFFFu)  // global_addr[56:32]
          | (2u << 30);                           // type=2 ("image")

    v8i g1;
    g1[0] = (int)(2u << 16);                      // data_size=4B; wg_mask=0
    g1[1] = 0;                                    // barrier=0; tdim0[15:0]=0
    g1[2] = 0x10;                                 // tdim0=1<<20; tdim1[15:0]=0
    g1[3] = (16 << 16) | 0x10;                    // tile_dim0=16; tdim1=1<<20
    g1[4] = (hcount << 16) | wcount;              // tile_dim2 | tile_dim1
    g1[5] = DT;                                   // dim0_stride = 1280
    g1[6] = (int)(((Ww * DT) & 0xFFFF) << 16);    // dim0_str[47:32]=0 | dim1_str[15:0]
    g1[7] = (Ww * DT) >> 16;                      // dim1_stride[47:16]

    v4i g2;
    g2[0] = 1 << 20;                              // tensor_dim2 (no clip)
    g2[1] = 1;                                    // tensor_dim3 (unused)
    g2[2] = 0;
    g2[3] = 0;                                    // tile_dim3 = 0
    v4i g3 = {0, 0, 0, 0};
    v8i g4 = {0, 0, 0, 0, 0, 0, 0, 0};            // clang-23 6-arg form
    __builtin_amdgcn_tensor_load_to_lds(g0, g1, g2, g3, g4, 0);
}
#endif

} // namespace

__global__ __launch_bounds__(NTHREADS)
void brain_gaussian_tdm_kernel(const float* __restrict__ img,
                               const float* __restrict__ noise,
                               float* __restrict__ out) {
    // double-buffered packed windows [hcount][wcount][16]: 2 x 30976 B
    __shared__ __align__(16) float tileb[2][TIH * TIW * DTC];
    // horizontally blurred rows [hcount][16][16]: 22528 B
    __shared__ __align__(16) float hbuf[TIH * TW * DTC];
    __shared__ int wmap[TIW];     // window w-pos -> packed col (reflect-101)
    __shared__ int hmap[TIH];     // window h-pos -> packed row (reflect-101)

    const float kc[7] = {0.004433048f, 0.054005582f, 0.242036223f, 0.399050281f,
                         0.242036223f, 0.054005582f, 0.004433048f};

    const int dtg = blockIdx.x;        // 20 groups of NCHUNK dt chunks
    const int wt  = blockIdx.y;        // 6 w tiles
    const int ht  = blockIdx.z % 6;    // 6 h tiles
    const int c   = blockIdx.z / 6;    // 2 channels

    const int tid = threadIdx.x;
    const int h0 = ht * TH, w0 = wt * TW;
    const int dt_base = dtg * (NCHUNK * DTC);

    // clipped valid window (reflected halo coords always land inside it)
    const int gh_s = max(h0 - 3, 0), gh_e = min(h0 + TH + 3, Hh);
    const int gw_s = max(w0 - 3, 0), gw_e = min(w0 + TW + 3, Ww);
    const int hcount = gh_e - gh_s;    // 19..22
    const int wcount = gw_e - gw_s;    // 19..22

    const size_t cbase = (size_t)c * Hh * Ww * DT;
    const float* win0 = img + cbase + ((size_t)gh_s * Ww + gw_s) * DT + dt_base;

    if (tid < TIW) wmap[tid] = refl101(w0 - 3 + tid, Ww) - gw_s;
    if (tid >= 32 && tid < 32 + TIH)
        hmap[tid - 32] = refl101(h0 - 3 + (tid - 32), Hh) - gh_s;

#if __has_builtin(__builtin_amdgcn_tensor_load_to_lds)
    if (tid < 32)   // wave 0 drives the DMA pipeline (TENSORcnt is per-wave)
        tdm_issue(win0, (uint32_t)(uintptr_t)tileb[0], wcount, hcount);
#endif

    for (int i = 0; i < NCHUNK; ++i) {
        const int dt0 = dt_base + i * DTC;
        const float* buf = tileb[i & 1];

#if __has_builtin(__builtin_amdgcn_tensor_load_to_lds)
        if (tid < 32) {
            if (i + 1 < NCHUNK) {   // prefetch next chunk into the other buffer
                tdm_issue(win0 + (i + 1) * DTC,
                          (uint32_t)(uintptr_t)tileb[(i + 1) & 1],
                          wcount, hcount);
                __builtin_amdgcn_s_wait_tensorcnt(1);  // chunk i landed
            } else {
                __builtin_amdgcn_s_wait_tensorcnt(0);  // last chunk landed
            }
        }
#else
        // fallback: per-lane async global->LDS DMA (ASYNCcnt), single buffer
        {
            const int nseg = hcount * wcount * (DTC / 4);
            for (int s = tid; s < nseg; s += NTHREADS) {
                const int seg = s & 3;
                const int pix = s >> 2;
                const int pw = pix % wcount;
                const int ph = pix / wcount;
                const float* gp = win0 + i * DTC + (size_t)ph * (Ww * DT)
                                  + (size_t)pw * DT + seg * 4;
                const uint32_t lp = (uint32_t)(uintptr_t)(
                    tileb[i & 1] + (ph * wcount + pw) * DTC + seg * 4);
                asm volatile("global_load_async_to_lds_b128 %0, %1, off"
                             :: "v"(lp), "v"(gp) : "memory");
            }
            asm volatile("s_wait_asynccnt 0x0" ::: "memory");
        }
#endif
        __syncthreads();   // buf[i&1] visible to all waves; hbuf free again

        // ---- horizontal 7-tap blur (w), reflection via wmap, float4 ----
        const int nhb4 = hcount * TW * (DTC / 4);
        for (int k = tid; k < nhb4; k += NTHREADS) {
            const int q  = k & 3;
            const int r  = k >> 2;
            const int tw = r % TW;
            const int ph = r / TW;
            const float* rowbase = buf + ph * wcount * DTC + q * 4;
            v4f acc = {0.f, 0.f, 0.f, 0.f};
#pragma unroll
            for (int j = 0; j < 7; ++j) {
                const v4f s = *reinterpret_cast<const v4f*>(rowbase + wmap[tw + j] * DTC);
                acc += kc[j] * s;
            }
            *reinterpret_cast<v4f*>(hbuf + (ph * TW + tw) * DTC + q * 4) = acc;
        }
        __syncthreads();

        // ---- vertical 7-tap blur (h) via hmap + fused noise, float4 ----
        for (int k = tid; k < TH * TW * (DTC / 4); k += NTHREADS) {  // 4 iters
            const int q  = k & 3;
            const int r  = k >> 2;
            const int tw = r % TW;
            const int th = r / TW;
            v4f acc = {0.f, 0.f, 0.f, 0.f};
#pragma unroll
            for (int j = 0; j < 7; ++j) {
                const v4f s = *reinterpret_cast<const v4f*>(
                    hbuf + (hmap[th + j] * TW + tw) * DTC + q * 4);
                acc += kc[j] * s;
            }
            const size_t o = cbase + ((size_t)(h0 + th) * Ww + (w0 + tw)) * DT
                             + dt0 + q * 4;
            const v4f nz = *reinterpret_cast<const v4f*>(noise + o);
            *reinterpret_cast<v4f*>(out + o) = acc + 0.05f * nz;
        }
        __syncthreads();   // hbuf reads done; buf[(i+2)&1] reads long done
    }
}

extern "C" void kernel_launch(void* const* d_in, const int* in_sizes, int n_in,
                              void* d_out, int out_size, void* d_ws, size_t ws_size,
                              hipStream_t stream) {
    const float* img   = (const float*)d_in[0];
    const float* noise = (const float*)d_in[1];
    float* out = (float*)d_out;

    dim3 grid(DT / (DTC * NCHUNK), Ww / TW, (Hh / TH) * 2);  // (20, 6, 12)
    brain_gaussian_tdm_kernel<<<grid, NTHREADS, 0, stream>>>(img, noise, out);
}